// DHDN_Dynamic_25065429139886
// MI455X (gfx1250) — compile-verified
//
#include <hip/hip_runtime.h>
#include <hip/hip_bf16.h>

typedef __attribute__((ext_vector_type(2))) float v2f;
typedef __attribute__((ext_vector_type(8))) float v8f;

#define Bq 4096
#define Jq 19
#define Hq 256

// ---------------------------------------------------------------------------
// fp32 WMMA GEMM:  C[M,N] = A[M,K] * W[N,K]^T (+ bias) (+ relu)
// One wave computes a 32x64 tile (2 m-tiles x 4 n-tiles) with
// V_WMMA_F32_16X16X4_F32, software-pipelined fragment loads (double buffer).
// Requires: M % 64 == 0 (2 waves/block, 32 rows each), K % 4 == 0.
// grid.x = M/64, grid.y = ceil(N/64)
// ---------------------------------------------------------------------------
__global__ __launch_bounds__(64) void gemm_wmma_f32(
    const float* __restrict__ A, const float* __restrict__ W,
    const float* __restrict__ bias, float* __restrict__ C,
    int M, int N, int K, int relu)
{
    const int lane = threadIdx.x & 31;
    const int wave = threadIdx.x >> 5;
    const int m0 = (blockIdx.x * 2 + wave) * 32;
    const int n0 = blockIdx.y * 64;
    if (m0 >= M) return;

    const int r  = lane & 15;   // row-within-16 / column-within-16
    const int hq = lane >> 4;   // half selector -> K offset 2*hq

    long arow[2];
#pragma unroll
    for (int m = 0; m < 2; ++m)
        arow[m] = (long)(m0 + 16 * m + r) * K + 2 * hq;

    long brow[4];
#pragma unroll
    for (int t = 0; t < 4; ++t) {
        int n = n0 + t * 16 + r;
        if (n >= N) n = N - 1;              // clamp; garbage cols masked at store
        brow[t] = (long)n * K + 2 * hq;
    }

    v8f acc[2][4];
#pragma unroll
    for (int m = 0; m < 2; ++m)
#pragma unroll
        for (int t = 0; t < 4; ++t)
#pragma unroll
            for (int v = 0; v < 8; ++v) acc[m][t][v] = 0.0f;

    // prologue: fragments for k = 0
    v2f aC[2], bC[4];
#pragma unroll
    for (int m = 0; m < 2; ++m) aC[m] = *reinterpret_cast<const v2f*>(A + arow[m]);
#pragma unroll
    for (int t = 0; t < 4; ++t) bC[t] = *reinterpret_cast<const v2f*>(W + brow[t]);

    int k = 0;
    for (; k + 4 < K; k += 4) {
        // issue next-iteration loads first so WMMA latency covers them
        v2f aN[2], bN[4];
#pragma unroll
        for (int m = 0; m < 2; ++m)
            aN[m] = *reinterpret_cast<const v2f*>(A + arow[m] + k + 4);
#pragma unroll
        for (int t = 0; t < 4; ++t)
            bN[t] = *reinterpret_cast<const v2f*>(W + brow[t] + k + 4);

#pragma unroll
        for (int m = 0; m < 2; ++m)
#pragma unroll
            for (int t = 0; t < 4; ++t)
                acc[m][t] = __builtin_amdgcn_wmma_f32_16x16x4_f32(
                    false, aC[m], false, bC[t], (short)0, acc[m][t], false, false);

#pragma unroll
        for (int m = 0; m < 2; ++m) aC[m] = aN[m];
#pragma unroll
        for (int t = 0; t < 4; ++t) bC[t] = bN[t];
    }
    // epilogue: last k-step
#pragma unroll
    for (int m = 0; m < 2; ++m)
#pragma unroll
        for (int t = 0; t < 4; ++t)
            acc[m][t] = __builtin_amdgcn_wmma_f32_16x16x4_f32(
                false, aC[m], false, bC[t], (short)0, acc[m][t], false, false);

#pragma unroll
    for (int m = 0; m < 2; ++m)
#pragma unroll
        for (int t = 0; t < 4; ++t) {
            const int col = n0 + t * 16 + r;
            if (col < N) {
                const float bv = bias ? bias[col] : 0.0f;
#pragma unroll
                for (int v = 0; v < 8; ++v) {
                    const int row = m0 + 16 * m + v + 8 * hq;  // C/D: vgpr v, hi-lanes -> M+8
                    float x = acc[m][t][v] + bv;
                    if (relu) x = fmaxf(x, 0.0f);
                    C[(long)row * N + col] = x;
                }
            }
        }
}

// ---------------------------------------------------------------------------
// Graph build: top-8 neighbours of 0.5*cdist(points)+0.5*cdist(feats).
// softmax is monotone in -dist => top_k(attn) == argmin-k(dist), ties->lowest idx.
// One 64-thread block per batch.
// ---------------------------------------------------------------------------
__global__ __launch_bounds__(64) void build_graph(
    const float* __restrict__ pts, const float* __restrict__ feats,
    const int* __restrict__ kv, int* __restrict__ idx, float* __restrict__ mask)
{
    const int b = blockIdx.x, t = threadIdx.x;
    __shared__ float sf[Jq * Hq];
    __shared__ float sp[Jq * 2];
    __shared__ float nf[Jq];
    for (int i = t; i < Jq * Hq; i += 64) sf[i] = feats[(long)b * Jq * Hq + i];
    for (int i = t; i < Jq * 2;  i += 64) sp[i] = pts[(long)b * Jq * 2 + i];
    __syncthreads();
    if (t < Jq) {
        float s = 0.0f;
        const float* f = &sf[t * Hq];
        for (int c = 0; c < Hq; ++c) s += f[c] * f[c];
        nf[t] = s;
    }
    __syncthreads();
    if (t < Jq) {
        float dist[Jq];
        const float px = sp[2 * t], py = sp[2 * t + 1];
        const float np_ = px * px + py * py;
        const float* fi = &sf[t * Hq];
        for (int j = 0; j < Jq; ++j) {
            const float* fj = &sf[j * Hq];
            float dot = 0.0f;
            for (int c = 0; c < Hq; ++c) dot += fi[c] * fj[c];
            const float d2f = nf[t] + nf[j] - 2.0f * dot;
            const float qx = sp[2 * j], qy = sp[2 * j + 1];
            const float d2p = np_ + qx * qx + qy * qy - 2.0f * (px * qx + py * qy);
            dist[j] = 0.5f * sqrtf(fmaxf(d2p, 0.0f)) + 0.5f * sqrtf(fmaxf(d2f, 0.0f));
        }
        const int kvi = kv[b * Jq + t];
        for (int k = 0; k < 8; ++k) {
            int bj = 0; float bd = dist[0];
            for (int j = 1; j < Jq; ++j)
                if (dist[j] < bd) { bd = dist[j]; bj = j; }   // strict < : lowest-index ties
            idx[(b * Jq + t) * 8 + k]  = bj;
            mask[(b * Jq + t) * 8 + k] = (k < kvi) ? 1.0f : 0.0f;
            dist[bj] = 3.4e38f;
        }
    }
}

// ---------------------------------------------------------------------------
// h0 = points @ w_init^T + b_init     (one block per row, 256 threads)
// ---------------------------------------------------------------------------
__global__ __launch_bounds__(256) void init_embed(
    const float* __restrict__ pts, const float* __restrict__ w,
    const float* __restrict__ bi, float* __restrict__ h)
{
    const long row = blockIdx.x;
    const int c = threadIdx.x;
    const float p0 = pts[row * 2], p1 = pts[row * 2 + 1];
    h[row * Hq + c] = p0 * w[c * 2] + p1 * w[c * 2 + 1] + bi[c];
}

// ---------------------------------------------------------------------------
// hconv aggregation (after xt GEMM) + relu + LayerNorm. One block per batch.
// Each thread owns one channel column => scatter-add is race free.
// ---------------------------------------------------------------------------
__global__ __launch_bounds__(256) void hconv_ln(
    const float* __restrict__ xt, const int* __restrict__ idx,
    const float* __restrict__ mask, const float* __restrict__ bg,
    const float* __restrict__ g, const float* __restrict__ bn,
    float* __restrict__ hout)
{
    const int b = blockIdx.x, t = threadIdx.x;
    __shared__ float sxt[Jq * Hq];
    __shared__ float sme[Jq * Hq];
    __shared__ float sout[Jq * Hq];
    __shared__ int   sidx[Jq * 8];
    __shared__ float smk[Jq * 8];
    __shared__ float sD[Jq];
    __shared__ float sinv[Jq];
    const long base = (long)b * Jq * Hq;
    for (int i = 0; i < Jq; ++i) sxt[i * Hq + t] = xt[base + i * Hq + t];
    if (t < Jq * 8) { sidx[t] = idx[b * Jq * 8 + t]; smk[t] = mask[b * Jq * 8 + t]; }
    __syncthreads();
    if (t < Jq) {
        float s = 0.0f;
        for (int k = 0; k < 8; ++k) s += smk[t * 8 + k];
        sinv[t] = 1.0f / fmaxf(s, 1.0f);
        float d = 0.0f;
        for (int e = 0; e < Jq * 8; ++e) if (sidx[e] == t) d += smk[e];
        sD[t] = d;
    }
    __syncthreads();
    // m_e[i] = masked mean of gathered xt rows
    for (int i = 0; i < Jq; ++i) {
        float acc = 0.0f;
        for (int k = 0; k < 8; ++k) acc += smk[i * 8 + k] * sxt[sidx[i * 8 + k] * Hq + t];
        sme[i * Hq + t] = acc * sinv[i];
    }
    // scatter-add into sout (column-exclusive per thread -> no races)
    for (int i = 0; i < Jq; ++i) sout[i * Hq + t] = 0.0f;
    for (int i = 0; i < Jq; ++i) {
        const float me = sme[i * Hq + t];
        for (int k = 0; k < 8; ++k) {
            const int j = sidx[i * 8 + k];
            sout[j * Hq + t] += me * smk[i * 8 + k];
        }
    }
    // out * (D>0 ? 1/D : 0) + b, relu
    for (int i = 0; i < Jq; ++i) {
        const float d = sD[i];
        const float v = sout[i * Hq + t] * (d > 0.0f ? 1.0f / d : 0.0f) + bg[t];
        sout[i * Hq + t] = fmaxf(v, 0.0f);
    }
    __syncthreads();
    // LayerNorm per row (256 channels) using wave32 shuffles
    const int wave = t >> 5, lane = t & 31;
    for (int r = wave; r < Jq; r += 8) {
        float x[8]; float s = 0.0f;
#pragma unroll
        for (int q = 0; q < 8; ++q) { x[q] = sout[r * Hq + lane + q * 32]; s += x[q]; }
        for (int off = 16; off >= 1; off >>= 1) s += __shfl_xor(s, off, 32);
        const float mu = s * (1.0f / 256.0f);
        float vs = 0.0f;
#pragma unroll
        for (int q = 0; q < 8; ++q) { const float d = x[q] - mu; vs += d * d; }
        for (int off = 16; off >= 1; off >>= 1) vs += __shfl_xor(vs, off, 32);
        const float inv = rsqrtf(vs * (1.0f / 256.0f) + 1e-5f);
#pragma unroll
        for (int q = 0; q < 8; ++q) {
            const int c = lane + q * 32;
            hout[base + r * Hq + c] = (x[q] - mu) * inv * g[c] + bn[c];
        }
    }
}

// ---------------------------------------------------------------------------
// 4-head attention over 19 tokens. One block per batch, thread = (head, d).
// ---------------------------------------------------------------------------
__global__ __launch_bounds__(256) void attn_kernel(
    const float* __restrict__ qkv, float* __restrict__ o)
{
    const int b = blockIdx.x, t = threadIdx.x;
    const int head = t >> 6, d = t & 63;
    __shared__ float sq[Jq * Hq], sk[Jq * Hq], sv[Jq * Hq];
    __shared__ float satt[4 * Jq];
    const long base = (long)b * Jq * 768;
    for (int i = 0; i < Jq; ++i) {
        sq[i * Hq + t] = qkv[base + i * 768 + t];
        sk[i * Hq + t] = qkv[base + i * 768 + 256 + t];
        sv[i * Hq + t] = qkv[base + i * 768 + 512 + t];
    }
    __syncthreads();
    for (int i = 0; i < Jq; ++i) {
        if (d < Jq) {
            const float* qi = &sq[i * Hq + head * 64];
            const float* kj = &sk[d * Hq + head * 64];
            float s = 0.0f;
            for (int c = 0; c < 64; ++c) s += qi[c] * kj[c];
            satt[head * Jq + d] = s * 0.125f;     // 1/sqrt(64)
        }
        __syncthreads();
        float mx = -3.4e38f;
        for (int j = 0; j < Jq; ++j) mx = fmaxf(mx, satt[head * Jq + j]);
        float den = 0.0f, acc = 0.0f;
        for (int j = 0; j < Jq; ++j) {
            const float e = expf(satt[head * Jq + j] - mx);
            den += e;
            acc += e * sv[j * Hq + head * 64 + d];
        }
        o[(long)b * Jq * Hq + i * Hq + head * 64 + d] = acc / den;
        __syncthreads();
    }
}

// ---------------------------------------------------------------------------
// z = mu + eps * exp(0.5 * logvar)
// ---------------------------------------------------------------------------
__global__ __launch_bounds__(256) void reparam(
    const float* __restrict__ zp, const float* __restrict__ eps, float* __restrict__ z)
{
    const int t = blockIdx.x * 256 + threadIdx.x;   // Bq*64 total
    const int b = t >> 6, p = t & 63;
    z[t] = zp[b * 128 + p] + eps[t] * expf(0.5f * zp[b * 128 + 64 + p]);
}

// ---------------------------------------------------------------------------
extern "C" void kernel_launch(void* const* d_in, const int* in_sizes, int n_in,
                              void* d_out, int out_size, void* d_ws, size_t ws_size,
                              hipStream_t stream)
{
    const float* points = (const float*)d_in[0];
    const float* feats  = (const float*)d_in[1];
    const int*   kvals  = (const int*)  d_in[2];
    const float* eps    = (const float*)d_in[3];
    const float* w_init = (const float*)d_in[4];
    const float* b_init = (const float*)d_in[5];
    const float* w_gnn  = (const float*)d_in[6];
    const float* b_gnn  = (const float*)d_in[7];
    const float* ln_g   = (const float*)d_in[8];
    const float* ln_b   = (const float*)d_in[9];
    const float* w_qkv  = (const float*)d_in[10];
    const float* b_qkv  = (const float*)d_in[11];
    const float* w_ao   = (const float*)d_in[12];
    const float* b_ao   = (const float*)d_in[13];
    const float* w_e1   = (const float*)d_in[14];
    const float* b_e1   = (const float*)d_in[15];
    const float* w_e2   = (const float*)d_in[16];
    const float* b_e2   = (const float*)d_in[17];
    const float* w_dec  = (const float*)d_in[18];
    const float* b_dec  = (const float*)d_in[19];
    const float* w_r1   = (const float*)d_in[20];
    const float* b_r1   = (const float*)d_in[21];
    const float* w_r2   = (const float*)d_in[22];
    const float* b_r2   = (const float*)d_in[23];
    float* out = (float*)d_out;

    char* ws = (char*)d_ws;
    size_t off = 0;
    auto alloc = [&](size_t n) { size_t o = off; off += (n + 255) & ~(size_t)255; return o; };
    const size_t ROWS = (size_t)Bq * Jq;                  // 77824
    int*   idx   = (int*)  (ws + alloc(ROWS * 8 * 4));
    float* mask  = (float*)(ws + alloc(ROWS * 8 * 4));
    float* hA    = (float*)(ws + alloc(ROWS * Hq * 4));
    float* hB    = (float*)(ws + alloc(ROWS * Hq * 4));
    float* xt    = (float*)(ws + alloc(ROWS * Hq * 4));
    float* qkvb  = (float*)(ws + alloc(ROWS * 768 * 4));
    float* enc1  = (float*)(ws + alloc((size_t)Bq * 256 * 4));
    float* zp    = (float*)(ws + alloc((size_t)Bq * 128 * 4));
    float* zz    = (float*)(ws + alloc((size_t)Bq * 64 * 4));
    float* r1o   = (float*)(ws + alloc((size_t)Bq * 256 * 4));
    float* hr    = qkvb;   // qkv dead after attention; reuse for decoder output

    // 1) graph + initial embedding
    build_graph<<<Bq, 64, 0, stream>>>(points, feats, kvals, idx, mask);
    init_embed<<<(unsigned)ROWS, 256, 0, stream>>>(points, w_init, b_init, hA);

    // 2) 3 GNN layers: WMMA GEMM (xt = h @ W^T) then aggregate+LN
    float* hcur = hA; float* hnext = hB;
    for (int l = 0; l < 3; ++l) {
        gemm_wmma_f32<<<dim3(1216, 4), 64, 0, stream>>>(
            hcur, w_gnn + (size_t)l * Hq * Hq, nullptr, xt, (int)ROWS, Hq, Hq, 0);
        hconv_ln<<<Bq, 256, 0, stream>>>(
            xt, idx, mask, b_gnn + l * Hq, ln_g + l * Hq, ln_b + l * Hq, hnext);
        float* tmp = hcur; hcur = hnext; hnext = tmp;
    }
    // hcur == hB here

    // 3) attention block
    gemm_wmma_f32<<<dim3(1216, 12), 64, 0, stream>>>(hcur, w_qkv, b_qkv, qkvb,
                                                     (int)ROWS, 768, Hq, 0);
    attn_kernel<<<Bq, 256, 0, stream>>>(qkvb, xt);
    gemm_wmma_f32<<<dim3(1216, 4), 64, 0, stream>>>(xt, w_ao, b_ao, hA,
                                                    (int)ROWS, Hq, Hq, 0);

    // 4) VAE encoder: hf = hA viewed as (4096, 4864)
    gemm_wmma_f32<<<dim3(64, 4), 64, 0, stream>>>(hA, w_e1, b_e1, enc1,
                                                  Bq, 256, Jq * Hq, 1);
    gemm_wmma_f32<<<dim3(64, 2), 64, 0, stream>>>(enc1, w_e2, b_e2, zp,
                                                  Bq, 128, 256, 0);
    reparam<<<(Bq * 64) / 256, 256, 0, stream>>>(zp, eps, zz);

    // 5) decoder + reconstruction head
    gemm_wmma_f32<<<dim3(64, 76), 64, 0, stream>>>(zz, w_dec, b_dec, hr,
                                                   Bq, Jq * Hq, 64, 0);
    gemm_wmma_f32<<<dim3(64, 4), 64, 0, stream>>>(hr, w_r1, b_r1, r1o,
                                                  Bq, 256, Jq * Hq, 1);
    gemm_wmma_f32<<<dim3(64, 1), 64, 0, stream>>>(r1o, w_r2, b_r2, out,
                                                  Bq, 2 * Jq, 256, 0);
}